// Attention_group_27367531610597
// MI455X (gfx1250) — compile-verified
//
#include <hip/hip_runtime.h>

typedef __attribute__((ext_vector_type(2))) float v2f;
typedef __attribute__((ext_vector_type(8))) float v8f;

#define TB   8            // batches per gating block
#define ROWS (TB * 45)    // 360 (b,i) rows
#define RPAD 368          // 23 tiles * 16 rows
#define NT   23           // column tiles per block

// workspace layout (in floats)
#define WS_GSUM   0       // 45 floats: sum over batch of gate
#define WS_MASK   64      // 45 ints: expert id per channel, or -1
#define WS_CONSTS 128     // 22 experts * 8 floats: a,b,c,d,P,Q+bo
#define WS_GATE   512     // B*45 floats

__global__ void zero_gsum_kernel(float* ws) {
    if (threadIdx.x < 45) ws[WS_GSUM + threadIdx.x] = 0.f;
}

// ---------------------------------------------------------------------------
// Gating: temp^T = (gc_w|gc_b) @ [X|1]^T via V_WMMA_F32_16X16X4_F32 so the
// 21-way pool reduces in-register (one shfl_xor(16) per tile); MLP also via
// WMMA (16 rows = 8 batches x {mx,av}); softmax -> gate; gate-sum atomics.
// x tile staged with GLOBAL_LOAD_ASYNC_TO_LDS_B32 (ASYNCcnt-fenced).
// ---------------------------------------------------------------------------
__global__ void __launch_bounds__(256) gating_kernel(
    const float* __restrict__ x, const float* __restrict__ gc_w,
    const float* __restrict__ gc_b,
    const float* __restrict__ w1, const float* __restrict__ b1,
    const float* __restrict__ w2, const float* __restrict__ b2,
    float* __restrict__ ws)
{
    __shared__ float Xs[RPAD * 24];   // [row][K]: K=0..20 data, 21 = 1.0 (bias)
    __shared__ float Wm[32 * 24];     // [o][K]: gc_w, K=21 col = gc_b, else 0
    __shared__ float mxA[ROWS], avA[ROWS];
    __shared__ float H1in[16 * 48];   // MLP input rows, K=45 -> 1.0 (bias)
    __shared__ float B1m[48 * 32];    // [K][N]: w1^T + bias row
    __shared__ float A2m[16 * 28];    // tanh(h1) rows, K=25 -> 1.0 (bias)
    __shared__ float B2m[28 * 48];    // [K][N]: w2^T + bias row
    __shared__ float mo[TB][2][45];
    __shared__ float gsum[45];

    const int tid = threadIdx.x;
    const int b0  = blockIdx.x * TB;

    // ---- async global->LDS copy of the x tile (pitched 21 -> 24 floats) ----
    const float* xb = x + (size_t)b0 * 945;
    const unsigned xs_base = (unsigned)(unsigned long long)(void*)Xs;
    for (int f = tid; f < TB * 945; f += 256) {
        const int row = f / 21, col = f % 21;
        const unsigned      lds_off = xs_base + (unsigned)(row * 24 + col) * 4u;
        const unsigned long long ga = (unsigned long long)(const void*)(xb + f);
        asm volatile("global_load_async_to_lds_b32 %0, %1, off"
                     :: "v"(lds_off), "v"(ga) : "memory");
    }

    // ---- descriptor fills (disjoint from async region) ----
    for (int f = tid; f < 32 * 24; f += 256) {
        int o = f / 24, k = f % 24;
        float v = 0.f;
        if (o < 21) { if (k < 21) v = gc_w[o * 21 + k]; else if (k == 21) v = gc_b[o]; }
        Wm[f] = v;
    }
    for (int f = tid; f < 48 * 32; f += 256) {
        int K = f >> 5, N = f & 31;
        float v = 0.f;
        if (N < 25) { if (K < 45) v = w1[N * 45 + K]; else if (K == 45) v = b1[N]; }
        B1m[f] = v;
    }
    for (int f = tid; f < 28 * 48; f += 256) {
        int K = f / 48, N = f % 48;
        float v = 0.f;
        if (N < 45) { if (K < 25) v = w2[N * 25 + K]; else if (K == 25) v = b2[N]; }
        B2m[f] = v;
    }
    // pad columns: K=21 -> 1.0 (bias), K=22..23 -> 0; pad rows 360..367 -> 0
    for (int f = tid; f < RPAD * 3; f += 256) {
        int r = f / 3, c2 = 21 + f % 3;
        Xs[r * 24 + c2] = (c2 == 21) ? 1.f : 0.f;
    }
    for (int f = tid; f < (RPAD - ROWS) * 21; f += 256) {
        int r = ROWS + f / 21;
        Xs[r * 24 + f % 21] = 0.f;
    }
    if (tid < 45) gsum[tid] = 0.f;

    asm volatile("s_wait_asynccnt 0x0" ::: "memory");
    __syncthreads();

    const int wave  = tid >> 5, lane = tid & 31;
    const int laneM = lane & 15, half = lane >> 4;
    const int laneN = laneM;

    // ---- gating GEMM: A = Wm (two o-tiles), B = Xs column tile ----
    for (int t = wave; t < NT; t += 8) {
        const int m0c = t * 16;
        v8f acc0 = {0.f,0.f,0.f,0.f,0.f,0.f,0.f,0.f};
        v8f acc1 = {0.f,0.f,0.f,0.f,0.f,0.f,0.f,0.f};
        const float* xr = &Xs[(m0c + laneN) * 24];
#pragma unroll
        for (int kk = 0; kk < 6; ++kk) {
            const int k0 = kk * 4 + 2 * half;
            v2f a0, a1, bv;
            a0.x = Wm[laneM * 24 + k0];        a0.y = Wm[laneM * 24 + k0 + 1];
            a1.x = Wm[(16 + laneM) * 24 + k0]; a1.y = Wm[(16 + laneM) * 24 + k0 + 1];
            bv.x = xr[k0];                     bv.y = xr[k0 + 1];
            acc0 = __builtin_amdgcn_wmma_f32_16x16x4_f32(
                       false, a0, false, bv, (short)0, acc0, false, false);
            acc1 = __builtin_amdgcn_wmma_f32_16x16x4_f32(
                       false, a1, false, bv, (short)0, acc1, false, false);
        }
        // half0 lane holds o=0..7 (+16..20), half1 holds o=8..15, col = laneN
        float m0s = acc0[0], s0 = acc0[0];
#pragma unroll
        for (int v = 1; v < 8; ++v) { m0s = fmaxf(m0s, acc0[v]); s0 += acc0[v]; }
        float m1 = acc1[0], s1 = acc1[0];
#pragma unroll
        for (int v = 1; v < 5; ++v) { m1 = fmaxf(m1, acc1[v]); s1 += acc1[v]; }
        float lm = (half == 0) ? fmaxf(m0s, m1) : m0s;
        float ls = (half == 0) ? (s0 + s1) : s0;
        float om = fmaxf(lm, __shfl_xor(lm, 16, 32));
        float os = ls + __shfl_xor(ls, 16, 32);
        const int col = m0c + laneN;
        if (half == 0 && col < ROWS) {
            mxA[col] = om;
            avA[col] = os * (1.f / 21.f);
        }
    }
    __syncthreads();

    // ---- MLP input: 16 rows = (batch, {mx,av}) ----
    for (int f = tid; f < 16 * 48; f += 256) {
        int r = f / 48, k = f % 48;
        float v = 0.f;
        if (k < 45)      v = (r & 1) ? avA[(r >> 1) * 45 + k] : mxA[(r >> 1) * 45 + k];
        else if (k == 45) v = 1.f;
        H1in[f] = v;
    }
    __syncthreads();

    if (wave == 0) {
        // GEMM1: 16x48 @ 48x32 (cols 0..24 valid)
        v8f c0 = {0.f,0.f,0.f,0.f,0.f,0.f,0.f,0.f};
        v8f c1 = {0.f,0.f,0.f,0.f,0.f,0.f,0.f,0.f};
#pragma unroll
        for (int kk = 0; kk < 12; ++kk) {
            const int k0 = kk * 4 + 2 * half;
            v2f a, bv0, bv1;
            a.x   = H1in[laneM * 48 + k0];    a.y   = H1in[laneM * 48 + k0 + 1];
            bv0.x = B1m[k0 * 32 + laneN];     bv0.y = B1m[(k0 + 1) * 32 + laneN];
            bv1.x = B1m[k0 * 32 + 16 + laneN];bv1.y = B1m[(k0 + 1) * 32 + 16 + laneN];
            c0 = __builtin_amdgcn_wmma_f32_16x16x4_f32(
                     false, a, false, bv0, (short)0, c0, false, false);
            c1 = __builtin_amdgcn_wmma_f32_16x16x4_f32(
                     false, a, false, bv1, (short)0, c1, false, false);
        }
        // tanh + transpose into A-layout for GEMM2 (pitch 28, K=25 bias col)
#pragma unroll
        for (int v = 0; v < 8; ++v) {
            const int row = v + 8 * half;
            A2m[row * 28 + laneN] = tanhf(c0[v]);
            if (laneN < 9)  A2m[row * 28 + 16 + laneN] = tanhf(c1[v]);
            if (laneN < 3)  A2m[row * 28 + 25 + laneN] = (laneN == 0) ? 1.f : 0.f;
        }
        // GEMM2: 16x28 @ 28x48 (cols 0..44 valid)
        v8f d0 = {0.f,0.f,0.f,0.f,0.f,0.f,0.f,0.f};
        v8f d1 = {0.f,0.f,0.f,0.f,0.f,0.f,0.f,0.f};
        v8f d2 = {0.f,0.f,0.f,0.f,0.f,0.f,0.f,0.f};
#pragma unroll
        for (int kk = 0; kk < 7; ++kk) {
            const int k0 = kk * 4 + 2 * half;
            v2f a, e0, e1, e2;
            a.x  = A2m[laneM * 28 + k0];       a.y  = A2m[laneM * 28 + k0 + 1];
            e0.x = B2m[k0 * 48 + laneN];       e0.y = B2m[(k0 + 1) * 48 + laneN];
            e1.x = B2m[k0 * 48 + 16 + laneN];  e1.y = B2m[(k0 + 1) * 48 + 16 + laneN];
            e2.x = B2m[k0 * 48 + 32 + laneN];  e2.y = B2m[(k0 + 1) * 48 + 32 + laneN];
            d0 = __builtin_amdgcn_wmma_f32_16x16x4_f32(
                     false, a, false, e0, (short)0, d0, false, false);
            d1 = __builtin_amdgcn_wmma_f32_16x16x4_f32(
                     false, a, false, e1, (short)0, d1, false, false);
            d2 = __builtin_amdgcn_wmma_f32_16x16x4_f32(
                     false, a, false, e2, (short)0, d2, false, false);
        }
#pragma unroll
        for (int v = 0; v < 8; ++v) {
            const int row = v + 8 * half;
            const int bb = row >> 1, wh = row & 1;
            mo[bb][wh][laneN]      = tanhf(d0[v]);
            mo[bb][wh][16 + laneN] = tanhf(d1[v]);
            if (laneN < 13) mo[bb][wh][32 + laneN] = tanhf(d2[v]);
        }
    }
    __syncthreads();

    // softmax over channels -> gate; block-local gate-sum
    if (tid < TB) {
        const int bb = tid;
        float sv[45], mx = -1e30f;
        for (int i = 0; i < 45; ++i) {
            sv[i] = mo[bb][0][i] + mo[bb][1][i];
            mx = fmaxf(mx, sv[i]);
        }
        float S = 0.f;
        for (int i = 0; i < 45; ++i) { sv[i] = expf(sv[i] - mx); S += sv[i]; }
        const float inv = 1.f / S;
        float* gout = ws + WS_GATE + (size_t)(b0 + bb) * 45;
        for (int i = 0; i < 45; ++i) {
            float g = sv[i] * inv;
            gout[i] = g;
            atomicAdd(&gsum[i], g);
        }
    }
    __syncthreads();
    if (tid < 45) atomicAdd(&ws[WS_GSUM + tid], gsum[tid]);
}

// ---------------------------------------------------------------------------
// Top-22 selection (stable rank == lax.top_k tie-break) + per-expert scalars
// ---------------------------------------------------------------------------
__global__ void select_kernel(
    const float* __restrict__ wq, const float* __restrict__ bq,
    const float* __restrict__ wk, const float* __restrict__ bk,
    const float* __restrict__ wv, const float* __restrict__ bv,
    const float* __restrict__ wo, const float* __restrict__ bo,
    float* __restrict__ ws)
{
    __shared__ float gs[45];
    __shared__ int   selF[45];
    const int t = threadIdx.x;
    if (t < 45) gs[t] = ws[WS_GSUM + t];
    __syncthreads();
    if (t < 45) {
        int rank = 0;
        for (int j = 0; j < 45; ++j)
            if (gs[j] > gs[t] || (gs[j] == gs[t] && j < t)) rank++;
        selF[t] = (rank < 22) ? 1 : 0;
    }
    __syncthreads();
    int* mask = (int*)(ws + WS_MASK);
    if (t < 45) {
        if (selF[t]) {
            int e = 0;
            for (int j = 0; j < t; ++j) e += selF[j];
            mask[t] = e;           // experts assigned in ascending channel order
        } else {
            mask[t] = -1;
        }
    }
    if (t < 22) {
        float a = 0, b = 0, c = 0, d = 0, P = 0, Q = 0;
        for (int i = 0; i < 21; ++i) {
            float q = wq[t * 21 + i], qb = bq[t * 21 + i];
            float k = wk[t * 21 + i], kb = bk[t * 21 + i];
            float v = wv[t * 21 + i], vb = bv[t * 21 + i];
            float o = wo[t * 21 + i];
            a += q * k;  b += q * kb;  c += qb * k;
            d += qb * kb; P += o * v;  Q += o * vb;
        }
        float* cst = ws + WS_CONSTS + t * 8;
        cst[0] = a; cst[1] = b; cst[2] = c;
        cst[3] = d; cst[4] = P; cst[5] = Q + bo[t];
    }
}

// ---------------------------------------------------------------------------
// Collapsed attention: one wave per (batch, channel). Unselected -> zeros.
// ---------------------------------------------------------------------------
__global__ void __launch_bounds__(256) attn_kernel(
    const float* __restrict__ x, const float* __restrict__ gate,
    const int* __restrict__ mask, const float* __restrict__ consts,
    float* __restrict__ G, float* __restrict__ A)
{
    const int  w    = threadIdx.x >> 5, lane = threadIdx.x & 31;
    const long p    = (long)blockIdx.x * 8 + w;
    const int  b    = (int)(p / 45), c = (int)(p % 45);
    const size_t base = (size_t)b * 945 + (size_t)c * 21;
    const int  e    = mask[c];                       // wave-uniform
    if (e < 0) {
        if (lane < 21) { G[base + lane] = 0.f; A[base + lane] = 0.f; }
        return;
    }
    const float* cst = consts + e * 8;
    const float ca = cst[0], cb = cst[1], cc = cst[2];
    const float cd = cst[3], cP = cst[4], cQ = cst[5];

    const float xv    = (lane < 21) ? x[base + lane] : 0.f;
    const float axl   = ca * xv;                     // a*x_l
    const float bxl_d = fmaf(cb, xv, cd);            // b*x_l + d
    float rmax = -1e30f;
#pragma unroll
    for (int m = 0; m < 21; ++m) {
        float xm = __shfl(xv, m, 32);
        float Em = fmaf(axl, xm, fmaf(cc, xm, bxl_d));
        rmax = fmaxf(rmax, Em);
    }
    float S = 0.f, Z = 0.f;
#pragma unroll
    for (int m = 0; m < 21; ++m) {
        float xm = __shfl(xv, m, 32);
        float Em = fmaf(axl, xm, fmaf(cc, xm, bxl_d));
        float ex = expf(Em - rmax);
        S += ex;
        Z = fmaf(ex, xm, Z);
    }
    const float y = fmaf(cP, Z / S, cQ);             // softmax rows sum to 1
    const float g = gate[(size_t)b * 45 + c];
    const float r = y * g;
    if (lane < 21) {
        A[base + lane] = r;
        G[base + lane] = xv * r;
    }
}

extern "C" void kernel_launch(void* const* d_in, const int* in_sizes, int n_in,
                              void* d_out, int out_size, void* d_ws, size_t ws_size,
                              hipStream_t stream)
{
    const float* x    = (const float*)d_in[0];
    const float* gc_w = (const float*)d_in[1];
    const float* gc_b = (const float*)d_in[2];
    const float* w1   = (const float*)d_in[3];
    const float* b1   = (const float*)d_in[4];
    const float* w2   = (const float*)d_in[5];
    const float* b2   = (const float*)d_in[6];
    const float* wq   = (const float*)d_in[7];
    const float* bq   = (const float*)d_in[8];
    const float* wk   = (const float*)d_in[9];
    const float* bk   = (const float*)d_in[10];
    const float* wv   = (const float*)d_in[11];
    const float* bv   = (const float*)d_in[12];
    const float* wo   = (const float*)d_in[13];
    const float* bo   = (const float*)d_in[14];

    float* ws = (float*)d_ws;
    float* G  = (float*)d_out;
    float* A  = G + (size_t)8192 * 945;

    zero_gsum_kernel<<<1, 64, 0, stream>>>(ws);
    gating_kernel<<<8192 / TB, 256, 0, stream>>>(x, gc_w, gc_b, w1, b1, w2, b2, ws);
    select_kernel<<<1, 64, 0, stream>>>(wq, bq, wk, bk, wv, bv, wo, bo, ws);
    attn_kernel<<<(8192 * 45) / 8, 256, 0, stream>>>(
        x, ws + WS_GATE, (const int*)(ws + WS_MASK), ws + WS_CONSTS, G, A);
}